// DFFN_41394894799832
// MI455X (gfx1250) — compile-verified
//
#include <hip/hip_runtime.h>
#include <math.h>

// ---------------------------------------------------------------------------
// DFFN on MI455X (gfx1250, wave32).
// Stages:
//   K0: build per-channel circulant kernels T_c from the real spectral filter
//   K1: project_in 1x1 conv as f32 WMMA GEMM (M=256, K=64), x tile staged via
//       GLOBAL_LOAD_ASYNC_TO_LDS_B128 (ASYNCcnt path, no VGPR round-trip).
//       LDS dest is passed to the asm as an addrspace(3) POINTER so the
//       shared buffer escapes and the compiler cannot fold its readers.
//   K2: spectral filter as per-channel 64x64 circulant GEMM (WMMA), in-place
//   K3: depthwise 3x3 + exact GELU gate + project_out GEMM (WMMA)
// Workspace: T (64KB) + intermediate y (4*256*256*256 f32 = 268.4MB)
// ---------------------------------------------------------------------------

typedef float v2f __attribute__((ext_vector_type(2)));
typedef float v8f __attribute__((ext_vector_type(8)));
typedef __attribute__((address_space(3))) float lds_f;

#define BATCH 4
#define CIN   64
#define C2    256
#define CHID  128
#define HH    256
#define WW    256
#define HWSZ  (HH * WW)

// f32 WMMA, D = A(16x4) * B(4x16) + C, fp32 throughout.
__device__ __forceinline__ v8f wmma_k4(v2f a, v2f b, v8f c) {
  return __builtin_amdgcn_wmma_f32_16x16x4_f32(false, a, false, b, (short)0, c,
                                               false, false);
}

// ---------------------------------------------------------------------------
// K0: T_c[dr][dc] = (1/64) * sum_{u,v in 0..7} G_c(u,v) * cos(2pi(u*dr+v*dc)/8)
// where G is the Hermitian extension of the (8 x 5) real rfft2 filter.
// irfft2(rfft2(x) * F) == circular-conv(x, T) per channel.
// ---------------------------------------------------------------------------
__global__ __launch_bounds__(256) void k_build_T(const float* __restrict__ filt,
                                                 float* __restrict__ T) {
  const int c = threadIdx.x;  // 256 channels, 1 block
  float F[40];
#pragma unroll
  for (int i = 0; i < 40; ++i) F[i] = filt[c * 40 + i];
  const float C8[8] = {1.f, 0.70710678118654752f, 0.f, -0.70710678118654752f,
                       -1.f, -0.70710678118654752f, 0.f, 0.70710678118654752f};
  float G[8][8];
  for (int u = 0; u < 8; ++u)
    for (int v = 0; v < 8; ++v)
      G[u][v] = (v <= 4) ? F[u * 5 + v] : F[((8 - u) & 7) * 5 + (8 - v)];
  for (int dr = 0; dr < 8; ++dr)
    for (int dc = 0; dc < 8; ++dc) {
      float s = 0.f;
      for (int u = 0; u < 8; ++u)
        for (int v = 0; v < 8; ++v)
          s += G[u][v] * C8[(u * dr + v * dc) & 7];
      T[c * 64 + dr * 8 + dc] = s * (1.0f / 64.0f);
    }
}

// ---------------------------------------------------------------------------
// K1: y[b, o, h, w] = sum_c w_in[o, c] * x[b, c, h, w]
// WG: one (b, h, 64-pixel strip). M=256 (16 tiles), N=64 (4 tiles), K=64.
// 8 waves; wave w owns M-tiles {2w, 2w+1} across all 4 N-tiles.
// x tile staged global->LDS with async B128 copies (ASYNCcnt).
// ---------------------------------------------------------------------------
#define BS_STR 72  // LDS row stride; 2*72 mod 64 = 16 -> conflict-free B reads
__global__ __launch_bounds__(256) void k_proj_in(const float* __restrict__ x,
                                                 const float* __restrict__ w_in,
                                                 float* __restrict__ y) {
  __shared__ float Bs[CIN * BS_STR];
  const int t = threadIdx.x;
  const int b = blockIdx.y;
  const int h = blockIdx.x >> 2;
  const int w0 = (blockIdx.x & 3) * 64;

  {  // stage x[b, 0..63, h, w0..w0+63] -> Bs[k][n] via async LDS copies.
    // Thread t owns row k = t>>2, 16 floats at n0 = (t&3)*16: four B128
    // copies; the offset: immediate advances global and LDS addresses
    // together, so one address pair covers all four. The AS(3) pointer
    // operand makes Bs escape into the asm (memory clobber then protects
    // the LDS reads below).
    const int k = t >> 2;
    const int n0 = (t & 3) * 16;
    const float* gsrc =
        x + ((size_t)b * CIN + k) * HWSZ + (size_t)h * WW + w0 + n0;
    lds_f* lp = (lds_f*)&Bs[k * BS_STR + n0];
    asm volatile(
        "global_load_async_to_lds_b128 %0, %1, off\n"
        "global_load_async_to_lds_b128 %0, %1, off offset:16\n"
        "global_load_async_to_lds_b128 %0, %1, off offset:32\n"
        "global_load_async_to_lds_b128 %0, %1, off offset:48\n"
        :
        : "v"(lp), "v"(gsrc)
        : "memory");
    asm volatile("s_wait_asynccnt 0" ::: "memory");
  }
  __syncthreads();

  const int wave = t >> 5, lane = t & 31;
  const int lm = lane & 15, lhi = lane >> 4;

  // A fragments (w_in rows) kept in registers across the N loop.
  v2f afr[2][16];
#pragma unroll
  for (int mt = 0; mt < 2; ++mt) {
    const float* wr = w_in + (wave * 32 + mt * 16 + lm) * CIN;
#pragma unroll
    for (int ks = 0; ks < 16; ++ks)
      afr[mt][ks] = *(const v2f*)(wr + ks * 4 + lhi * 2);
  }

  for (int nt = 0; nt < 4; ++nt) {
    v8f acc0 = {0, 0, 0, 0, 0, 0, 0, 0}, acc1 = acc0;
    const int n = nt * 16 + lm;
#pragma unroll
    for (int ks = 0; ks < 16; ++ks) {
      const int kb = ks * 4 + lhi * 2;
      v2f bf = {Bs[kb * BS_STR + n], Bs[(kb + 1) * BS_STR + n]};
      acc0 = wmma_k4(afr[0][ks], bf, acc0);
      acc1 = wmma_k4(afr[1][ks], bf, acc1);
    }
    const size_t col = (size_t)h * WW + w0 + n;
#pragma unroll
    for (int r = 0; r < 8; ++r) {
      const int ch0 = wave * 32 + r + 8 * lhi;
      y[((size_t)b * C2 + ch0) * HWSZ + col] = acc0[r];
      y[((size_t)b * C2 + ch0 + 16) * HWSZ + col] = acc1[r];
    }
  }
}

// ---------------------------------------------------------------------------
// K2: per (b, c, patch-row): Z[p, pw] = M_c[p, q] * Y[q, pw], 32 patches.
// M_c expanded in LDS from circulant T_c. In-place on y (WG owns its region;
// data read into LDS before overwrite). M=64 (4 tiles) x N=32 (2 tiles),
// one 16x16 tile per wave, K=64.
// ---------------------------------------------------------------------------
#define MS_STR 68  // A-read pattern (4m+kb banks) conflict-free
#define YS_STR 40  // 2*40 mod 64 = 16 -> conflict-free B reads
__global__ __launch_bounds__(256) void k_spectral(float* __restrict__ y,
                                                  const float* __restrict__ T) {
  __shared__ float Ms[64 * MS_STR];
  __shared__ float Ys[64 * YS_STR];
  const int t = threadIdx.x;
  const int ph = blockIdx.x;  // patch row 0..31
  const int c = blockIdx.y;   // channel 0..255
  const int b = blockIdx.z;
  const float* Tc = T + c * 64;

#pragma unroll
  for (int i = 0; i < 16; ++i) {  // expand circulant -> 64x64 matrix
    const int idx = t * 16 + i;
    const int p = idx >> 6, q = idx & 63;
    const int dr = ((p >> 3) - (q >> 3)) & 7;
    const int dc = ((p & 7) - (q & 7)) & 7;
    Ms[p * MS_STR + q] = Tc[dr * 8 + dc];
  }
  float* base = y + ((size_t)b * C2 + c) * HWSZ + (size_t)(ph * 8) * WW;
#pragma unroll
  for (int i = 0; i < 8; ++i) {  // gather 32 patches -> Ys[q][pw]
    const int lin = t * 8 + i;
    const int q = lin >> 5, pw = lin & 31;
    Ys[q * YS_STR + pw] = base[(size_t)(q >> 3) * WW + pw * 8 + (q & 7)];
  }
  __syncthreads();

  const int wave = t >> 5, lane = t & 31;
  const int lm = lane & 15, lhi = lane >> 4;
  const int mt = wave >> 1, nt = wave & 1;
  const int m = mt * 16 + lm;
  const int n = nt * 16 + lm;
  v8f acc = {0, 0, 0, 0, 0, 0, 0, 0};
#pragma unroll
  for (int ks = 0; ks < 16; ++ks) {
    const int kb = ks * 4 + lhi * 2;
    v2f af = {Ms[m * MS_STR + kb], Ms[m * MS_STR + kb + 1]};
    v2f bf = {Ys[kb * YS_STR + n], Ys[(kb + 1) * YS_STR + n]};
    acc = wmma_k4(af, bf, acc);
  }
#pragma unroll
  for (int r = 0; r < 8; ++r) {
    const int p = mt * 16 + r + 8 * lhi;
    base[(size_t)(p >> 3) * WW + n * 8 + (p & 7)] = acc[r];
  }
}

// ---------------------------------------------------------------------------
// K3: per 8x8 output tile: stage z halo (10x10 x 256ch) in dynamic LDS,
// depthwise 3x3 + exact-erf GELU gate -> g[128 x 64] in LDS, then
// out[64 x 64] = w_out[64x128] * g via WMMA (16 tiles, 2 per wave, K=128).
// ---------------------------------------------------------------------------
#define ZST 100
#define GST 72
__global__ __launch_bounds__(256) void k_tail(const float* __restrict__ z,
                                              const float* __restrict__ w_dw,
                                              const float* __restrict__ w_out,
                                              float* __restrict__ out) {
  extern __shared__ float smem[];
  float* zs = smem;            // C2 * ZST floats
  float* gs = smem + C2 * ZST; // CHID * GST floats
  const int t = threadIdx.x;
  const int h0 = blockIdx.y * 8, w0 = blockIdx.x * 8;
  const int b = blockIdx.z;

  for (int j = 0; j < 128; ++j) {  // stage 10x10 halo per channel (zero-pad)
    const int lin = t + 256 * j;
    const int c = lin >> 7, s = lin & 127;
    if (s < 100) {
      const int sy = s / 10, sx = s - sy * 10;
      const int gh = h0 + sy - 1, gw = w0 + sx - 1;
      float v = 0.0f;
      if (gh >= 0 && gh < HH && gw >= 0 && gw < WW)
        v = z[((size_t)b * C2 + c) * HWSZ + (size_t)gh * WW + gw];
      zs[c * ZST + s] = v;
    }
  }
  __syncthreads();

  for (int i = 0; i < 32; ++i) {  // depthwise 3x3 + GELU gate
    const int lin = t + 256 * i;
    const int cc = lin >> 6, pix = lin & 63;
    const int r = pix >> 3, cp = pix & 7;
    const float* k1 = w_dw + cc * 9;
    const float* k2 = w_dw + (cc + CHID) * 9;
    const float* z1 = zs + cc * ZST + r * 10 + cp;
    const float* z2 = zs + (cc + CHID) * ZST + r * 10 + cp;
    float d1 = 0.f, d2 = 0.f;
#pragma unroll
    for (int dy = 0; dy < 3; ++dy)
#pragma unroll
      for (int dx = 0; dx < 3; ++dx) {
        d1 += k1[dy * 3 + dx] * z1[dy * 10 + dx];
        d2 += k2[dy * 3 + dx] * z2[dy * 10 + dx];
      }
    const float g1 = 0.5f * d1 * (1.0f + erff(d1 * 0.70710678118654752f));
    gs[cc * GST + pix] = g1 * d2;
  }
  __syncthreads();

  const int wave = t >> 5, lane = t & 31;
  const int lm = lane & 15, lhi = lane >> 4;
  for (int tile = wave; tile < 16; tile += 8) {
    const int mt = tile >> 2, nt = tile & 3;
    const int m = mt * 16 + lm, n = nt * 16 + lm;
    const float* wr = w_out + m * CHID;
    v8f acc = {0, 0, 0, 0, 0, 0, 0, 0};
#pragma unroll
    for (int ks = 0; ks < 32; ++ks) {
      const int kb = ks * 4 + lhi * 2;
      v2f af = {wr[kb], wr[kb + 1]};
      v2f bf = {gs[kb * GST + n], gs[(kb + 1) * GST + n]};
      acc = wmma_k4(af, bf, acc);
    }
#pragma unroll
    for (int r = 0; r < 8; ++r) {
      const int o = mt * 16 + r + 8 * lhi;
      out[((size_t)b * CIN + o) * HWSZ + (size_t)(h0 + (n >> 3)) * WW + w0 +
          (n & 7)] = acc[r];
    }
  }
}

// ---------------------------------------------------------------------------
extern "C" void kernel_launch(void* const* d_in, const int* in_sizes, int n_in,
                              void* d_out, int out_size, void* d_ws,
                              size_t ws_size, hipStream_t stream) {
  (void)in_sizes; (void)n_in; (void)out_size; (void)ws_size;
  const float* x = (const float*)d_in[0];
  const float* w_in = (const float*)d_in[1];
  const float* w_dw = (const float*)d_in[2];
  const float* filt = (const float*)d_in[3];
  const float* w_out = (const float*)d_in[4];
  float* out = (float*)d_out;

  float* T = (float*)d_ws;            // 256*64 floats = 64KB
  float* y = (float*)d_ws + 16384;    // intermediate, 268.4MB

  hipLaunchKernelGGL(k_build_T, dim3(1), dim3(256), 0, stream, filt, T);
  hipLaunchKernelGGL(k_proj_in, dim3((HH * WW / 64), BATCH), dim3(256), 0,
                     stream, x, w_in, y);
  hipLaunchKernelGGL(k_spectral, dim3(32, C2, BATCH), dim3(256), 0, stream, y,
                     T);
  hipLaunchKernelGGL(k_tail, dim3(WW / 8, HH / 8, BATCH), dim3(256),
                     (C2 * ZST + CHID * GST) * sizeof(float), stream, y, w_dw,
                     w_out, out);
}